// TemporalEncoder_20272245637359
// MI455X (gfx1250) — compile-verified
//
#include <hip/hip_runtime.h>
#include <hip/hip_bf16.h>

#define HH      512
#define NHEAD   8
#define HD      64
#define TSTEPS  16
#define BB      8
#define HWN     3072
#define NKEYS   179
#define H4      2048
#define OUTN    49152

typedef __attribute__((ext_vector_type(2))) float v2f;
typedef __attribute__((ext_vector_type(8))) float v8f;

__device__ __forceinline__ float sigf(float x) { return 1.0f / (1.0f + __expf(-x)); }

// ---------------- WMMA fp32 16x16 tile GEMM core (1 tile / wave) ----------------
// A layout (ISA 7.12.2, 32-bit A 16x4): lanes 0-15 hold K=k,k+1; lanes 16-31 K=k+2,k+3.
// Rows >= M zero-masked via clamped addresses (no divergence; EXEC all-1s at WMMA).
__device__ __forceinline__ v8f wmma_kloop(const float* __restrict__ A, int K, int M,
                                          const float* __restrict__ W, int N,
                                          int row0, int col0, int lane, v8f acc) {
  const int half = lane >> 4;
  const int l    = lane & 15;
  const int ar   = row0 + l;
  const int arc  = (ar < M) ? ar : 0;
  const float am = (ar < M) ? 1.0f : 0.0f;
  const float* __restrict__ Arow = A + (long)arc * K;
  const float* __restrict__ Wcol = W + col0 + l;
  #pragma unroll 4
  for (int k = 0; k < K; k += 4) {
    const int kk = k + half * 2;
    v2f a, b;
    a[0] = Arow[kk] * am;
    a[1] = Arow[kk + 1] * am;
    b[0] = Wcol[(long)kk * N];
    b[1] = Wcol[(long)(kk + 1) * N];
    acc = __builtin_amdgcn_wmma_f32_16x16x4_f32(false, a, false, b, (short)0, acc,
                                                false, false);
  }
  return acc;
}

// D[MxN] = A1@W1 (+ A2@W2) (+bias) (+addm) (relu?)  -- small-M / latency path
__global__ __launch_bounds__(256)
void k_gemm(const float* __restrict__ A1, int K1, const float* __restrict__ W1,
            const float* __restrict__ A2, int K2, const float* __restrict__ W2,
            const float* __restrict__ bias, const float* __restrict__ addm,
            float* __restrict__ D, int M, int N, int relu) {
  const int wave = (blockIdx.x * blockDim.x + threadIdx.x) >> 5;
  const int lane = threadIdx.x & 31;
  const int tilesN = N >> 4;
  const int tilesM = (M + 15) >> 4;
  if (wave >= tilesM * tilesN) return;            // wave-uniform exit
  const int tm = wave / tilesN, tn = wave % tilesN;
  const int row0 = tm << 4, col0 = tn << 4;
  v8f acc = {};
  acc = wmma_kloop(A1, K1, M, W1, N, row0, col0, lane, acc);
  if (A2) acc = wmma_kloop(A2, K2, M, W2, N, row0, col0, lane, acc);
  const int half = lane >> 4, l = lane & 15;
  const int col = col0 + l;
  const float bb = bias ? bias[col] : 0.0f;
  #pragma unroll
  for (int r = 0; r < 8; ++r) {
    const int row = row0 + r + half * 8;
    if (row < M) {
      float v = acc[r] + bb;
      if (addm) v += addm[(long)row * N + col];
      if (relu) v = fmaxf(v, 0.0f);
      D[(long)row * N + col] = v;
    }
  }
}

// ---------------- 4-wide column-tiled WMMA GEMM (16x64 / wave) ----------------
// A fragment loaded once, reused by 4 WMMAs into 4 independent accumulators:
// A-traffic /4 and no same-register deps between consecutive v_wmma ops.
__global__ __launch_bounds__(256)
void k_gemm4(const float* __restrict__ A1, int K1, const float* __restrict__ W1,
             const float* __restrict__ bias, float* __restrict__ D,
             int M, int N, int relu) {
  const int wave = (blockIdx.x * blockDim.x + threadIdx.x) >> 5;
  const int lane = threadIdx.x & 31;
  const int tilesN4 = N >> 6;
  const int tilesM  = (M + 15) >> 4;
  if (wave >= tilesM * tilesN4) return;           // wave-uniform exit
  const int tm = wave / tilesN4, tn = wave % tilesN4;
  const int row0 = tm << 4, col0 = tn << 6;
  const int half = lane >> 4;
  const int l    = lane & 15;
  const int ar   = row0 + l;
  const int arc  = (ar < M) ? ar : 0;
  const float am = (ar < M) ? 1.0f : 0.0f;
  const float* __restrict__ Arow = A1 + (long)arc * K1;
  const float* __restrict__ Wc   = W1 + col0 + l;
  v8f acc0 = {}, acc1 = {}, acc2 = {}, acc3 = {};
  #pragma unroll 2
  for (int k = 0; k < K1; k += 4) {
    const int kk = k + half * 2;
    v2f a;
    a[0] = Arow[kk] * am;
    a[1] = Arow[kk + 1] * am;
    const float* __restrict__ W0 = Wc + (long)kk * N;
    const float* __restrict__ Wn = Wc + (long)(kk + 1) * N;
    v2f b0; b0[0] = W0[0];  b0[1] = Wn[0];
    v2f b1; b1[0] = W0[16]; b1[1] = Wn[16];
    v2f b2; b2[0] = W0[32]; b2[1] = Wn[32];
    v2f b3; b3[0] = W0[48]; b3[1] = Wn[48];
    acc0 = __builtin_amdgcn_wmma_f32_16x16x4_f32(false, a, false, b0, (short)0, acc0, false, false);
    acc1 = __builtin_amdgcn_wmma_f32_16x16x4_f32(false, a, false, b1, (short)0, acc1, false, false);
    acc2 = __builtin_amdgcn_wmma_f32_16x16x4_f32(false, a, false, b2, (short)0, acc2, false, false);
    acc3 = __builtin_amdgcn_wmma_f32_16x16x4_f32(false, a, false, b3, (short)0, acc3, false, false);
  }
  v8f accs[4] = {acc0, acc1, acc2, acc3};
  #pragma unroll
  for (int tte = 0; tte < 4; ++tte) {
    const int col = col0 + tte * 16 + l;
    const float bb = bias ? bias[col] : 0.0f;
    #pragma unroll
    for (int r = 0; r < 8; ++r) {
      const int row = row0 + r + half * 8;
      if (row < M) {
        float v = accs[tte][r] + bb;
        if (relu) v = fmaxf(v, 0.0f);
        D[(long)row * N + col] = v;
      }
    }
  }
}

// ---------------- setup kernels ----------------
__global__ void k_embed(const int* __restrict__ x, const int* __restrict__ y,
                        const int* __restrict__ lc, const int* __restrict__ rc,
                        const int* __restrict__ ke,
                        const float* __restrict__ ex, const float* __restrict__ ey,
                        const float* __restrict__ elc, const float* __restrict__ erc,
                        const float* __restrict__ ekey, float* __restrict__ E) {
  const int tb = blockIdx.x;
  __shared__ int ks[NKEYS];
  for (int k = threadIdx.x; k < NKEYS; k += blockDim.x)
    ks[k] = ke[tb * NKEYS + k] + 2 * k;
  __syncthreads();
  const int xi = x[tb], yi = y[tb], li = lc[tb], ri = rc[tb];
  float* e = E + (long)tb * 1536;
  for (int c = threadIdx.x; c < HH; c += blockDim.x) {
    e[c]       = ex[(long)xi * HH + c];
    e[HH + c]  = ey[(long)yi * HH + c];
    float s = elc[li * HH + c] + erc[ri * HH + c];
    for (int k = 0; k < NKEYS; ++k) s += ekey[(long)ks[k] * HH + c];
    e[1024 + c] = s;
  }
}

__global__ void k_fold(const float* __restrict__ Wimg, const float* __restrict__ Wk,
                       const float* __restrict__ Wv, float* __restrict__ Wk4,
                       float* __restrict__ Wv4) {
  const int i = blockIdx.x * blockDim.x + threadIdx.x;  // 0..2047
  if (i >= 4 * HH) return;
  const int j = i >> 9, c = i & 511;
  float sk = 0.f, sv = 0.f;
  for (int m = 0; m < HH; ++m) {
    const float w = Wimg[j * HH + m];
    sk += w * Wk[m * HH + c];
    sv += w * Wv[m * HH + c];
  }
  Wk4[i] = sk; Wv4[i] = sv;
}

__global__ void k_addbias(const float* __restrict__ pe, const float* __restrict__ bimg,
                          float* __restrict__ tmp) {
  const int i = blockIdx.x * blockDim.x + threadIdx.x;
  if (i < HWN * HH) tmp[i] = pe[i] + bimg[i & 511];
}

// ---------------- per-step state kernels ----------------
__global__ void k_init(const float* u_hlo, const float* u_clo, const float* u_hup,
                       const float* u_cup, const float* u_fb,
                       float* h_lo, float* c_lo, float* h_up, float* c_up, float* fb) {
  const int i = blockIdx.x * blockDim.x + threadIdx.x;
  if (i >= BB * HH) return;
  const int c = i & 511;
  h_lo[i] = u_hlo[c]; c_lo[i] = u_clo[c];
  h_up[i] = u_hup[c]; c_up[i] = u_cup[c]; fb[i] = u_fb[c];
}

__global__ void k_mask(const int* __restrict__ pad_t,
                       const float* p_hlo, const float* p_clo, const float* p_hup,
                       const float* p_cup, const float* p_fb,
                       float* h_lo, float* c_lo, float* h_up, float* c_up, float* fb) {
  const int i = blockIdx.x * blockDim.x + threadIdx.x;
  if (i >= BB * HH) return;
  const int b = i >> 9, c = i & 511;
  if (pad_t[b]) {
    h_lo[i] = p_hlo[c]; c_lo[i] = p_clo[c];
    h_up[i] = p_hup[c]; c_up[i] = p_cup[c]; fb[i] = p_fb[c];
  }
}

__global__ void k_lstm(const float* __restrict__ g, float* __restrict__ c,
                       float* __restrict__ h, float* __restrict__ fb) {
  const int i = blockIdx.x * blockDim.x + threadIdx.x;
  if (i >= BB * HH) return;
  const int b = i >> 9, cc = i & 511;
  const float* gb = g + (long)b * H4;
  const float gi = gb[cc], gf = gb[HH + cc], gg = gb[1024 + cc], go = gb[1536 + cc];
  const float cN = sigf(gf) * c[i] + sigf(gi) * tanhf(gg);
  const float hN = sigf(go) * tanhf(cN);
  c[i] = cN; h[i] = hN;
  if (fb) fb[i] = hN;
}

// ---------------- attention (folded K/V form) ----------------
__global__ __launch_bounds__(256)
void k_scores(const float* __restrict__ q, const float* __restrict__ img_t,
              const float* __restrict__ posk, const float* __restrict__ Wk4,
              float* __restrict__ scores) {
  const int bh = blockIdx.x;
  const int b = bh >> 3, h = bh & 7;
  __shared__ float qh[HD];
  __shared__ float qw[4];
  const int tid = threadIdx.x;
  if (tid < HD) qh[tid] = q[b * HH + h * HD + tid];
  __syncthreads();
  if (tid < 4) {
    float s = 0.f;
    for (int d = 0; d < HD; ++d) s += qh[d] * Wk4[tid * HH + h * HD + d];
    qw[tid] = s;
  }
  __syncthreads();
  const float* ib = img_t + (long)b * 4 * HWN;
  for (int s = tid; s < HWN; s += 256) {
    float dot = qw[0] * ib[s] + qw[1] * ib[HWN + s] + qw[2] * ib[2 * HWN + s] +
                qw[3] * ib[3 * HWN + s];
    const float* pk = posk + (long)s * HH + h * HD;
    float dp = 0.f;
    #pragma unroll 8
    for (int d = 0; d < HD; ++d) dp += qh[d] * pk[d];
    scores[(long)bh * HWN + s] = (dot + dp) * 0.125f;
  }
}

__global__ __launch_bounds__(256)
void k_softmax(float* __restrict__ scores) {
  float* sc = scores + (long)blockIdx.x * HWN;
  __shared__ float red[256];
  const int tid = threadIdx.x;
  float m = -1e30f;
  for (int s = tid; s < HWN; s += 256) m = fmaxf(m, sc[s]);
  red[tid] = m; __syncthreads();
  for (int w = 128; w > 0; w >>= 1) { if (tid < w) red[tid] = fmaxf(red[tid], red[tid + w]); __syncthreads(); }
  m = red[0]; __syncthreads();
  float sum = 0.f;
  for (int s = tid; s < HWN; s += 256) { const float e = __expf(sc[s] - m); sc[s] = e; sum += e; }
  red[tid] = sum; __syncthreads();
  for (int w = 128; w > 0; w >>= 1) { if (tid < w) red[tid] += red[tid + w]; __syncthreads(); }
  const float inv = 1.0f / red[0];
  for (int s = tid; s < HWN; s += 256) sc[s] *= inv;
}

__global__ __launch_bounds__(256)
void k_ctx(const float* __restrict__ attn, const float* __restrict__ img_t,
           const float* __restrict__ posv, const float* __restrict__ Wv4,
           float* __restrict__ ctx) {
  const int bh = blockIdx.x;
  const int b = bh >> 3, h = bh & 7;
  const int tid = threadIdx.x;
  const float* at = attn + (long)bh * HWN;
  __shared__ float red[256];
  __shared__ float aimg[4];
  for (int j = 0; j < 4; ++j) {
    float p = 0.f;
    const float* ib = img_t + ((long)b * 4 + j) * HWN;
    for (int s = tid; s < HWN; s += 256) p += at[s] * ib[s];
    red[tid] = p; __syncthreads();
    for (int w = 128; w > 0; w >>= 1) { if (tid < w) red[tid] += red[tid + w]; __syncthreads(); }
    if (tid == 0) aimg[j] = red[0];
    __syncthreads();
  }
  if (tid < HD) {
    float cv = 0.f;
    const float* pv = posv + h * HD + tid;
    for (int s = 0; s < HWN; ++s) cv += at[s] * pv[(long)s * HH];
    for (int j = 0; j < 4; ++j) cv += aimg[j] * Wv4[j * HH + h * HD + tid];
    ctx[b * HH + h * HD + tid] = cv;
  }
}

// ---------------- host launcher ----------------
extern "C" void kernel_launch(void* const* d_in, const int* in_sizes, int n_in,
                              void* d_out, int out_size, void* d_ws, size_t ws_size,
                              hipStream_t stream) {
  const float* imgf   = (const float*)d_in[0];
  const int*   xi     = (const int*)d_in[1];
  const int*   yi     = (const int*)d_in[2];
  const int*   lci    = (const int*)d_in[3];
  const int*   rci    = (const int*)d_in[4];
  const int*   kei    = (const int*)d_in[5];
  const int*   padi   = (const int*)d_in[6];
  const float* pad_h_lo = (const float*)d_in[7];
  const float* unk_h_lo = (const float*)d_in[8];
  const float* pad_h_up = (const float*)d_in[9];
  const float* unk_h_up = (const float*)d_in[10];
  const float* pad_c_lo = (const float*)d_in[11];
  const float* unk_c_lo = (const float*)d_in[12];
  const float* pad_c_up = (const float*)d_in[13];
  const float* unk_c_up = (const float*)d_in[14];
  const float* fb_pad   = (const float*)d_in[15];
  const float* fb_unk   = (const float*)d_in[16];
  const float* pos_emb  = (const float*)d_in[17];
  const float* Wimg     = (const float*)d_in[18];
  const float* bimg     = (const float*)d_in[19];
  const float* emb_x    = (const float*)d_in[20];
  const float* emb_y    = (const float*)d_in[21];
  const float* emb_lc   = (const float*)d_in[22];
  const float* emb_rc   = (const float*)d_in[23];
  const float* emb_key  = (const float*)d_in[24];
  const float* Wproj    = (const float*)d_in[25];
  const float* bproj    = (const float*)d_in[26];
  const float* Wq       = (const float*)d_in[27];
  const float* bq       = (const float*)d_in[28];
  const float* Wk       = (const float*)d_in[29];
  const float* bk       = (const float*)d_in[30];
  const float* Wv       = (const float*)d_in[31];
  const float* bv       = (const float*)d_in[32];
  const float* Wo       = (const float*)d_in[33];
  const float* bo       = (const float*)d_in[34];
  const float* Wih_lo   = (const float*)d_in[35];
  const float* Whh_lo   = (const float*)d_in[36];
  const float* b_lo     = (const float*)d_in[37];
  const float* Wih_up   = (const float*)d_in[38];
  const float* Whh_up   = (const float*)d_in[39];
  const float* b_up     = (const float*)d_in[40];
  const float* Wout1    = (const float*)d_in[41];
  const float* bout1    = (const float*)d_in[42];
  const float* Wout2    = (const float*)d_in[43];
  const float* bout2    = (const float*)d_in[44];

  float* ws = (float*)d_ws;
  size_t off = 0;
  float* E     = ws + off; off += (size_t)TSTEPS * BB * 1536;
  float* tmpPB = ws + off; off += (size_t)HWN * HH;
  float* posk  = ws + off; off += (size_t)HWN * HH;
  float* posv  = ws + off; off += (size_t)HWN * HH;
  float* Wk4   = ws + off; off += 4 * HH;
  float* Wv4   = ws + off; off += 4 * HH;
  float* P0    = ws + off; off += (size_t)TSTEPS * BB * H4;
  float* h_lo  = ws + off; off += BB * HH;
  float* c_lo  = ws + off; off += BB * HH;
  float* h_up  = ws + off; off += BB * HH;
  float* c_up  = ws + off; off += BB * HH;
  float* fb    = ws + off; off += BB * HH;
  float* proj  = ws + off; off += BB * H4;
  float* g     = ws + off; off += BB * H4;
  float* qb    = ws + off; off += BB * HH;
  float* sc    = ws + off; off += (size_t)BB * NHEAD * HWN;
  float* ctx   = ws + off; off += BB * HH;
  float* ctxo  = ws + off; off += BB * HH;
  float* hid   = ws + off; off += BB * H4;

  // ---- setup (all parallel across T) ----
  k_embed<<<TSTEPS * BB, 256, 0, stream>>>(xi, yi, lci, rci, kei, emb_x, emb_y,
                                           emb_lc, emb_rc, emb_key, E);
  k_fold<<<8, 256, 0, stream>>>(Wimg, Wk, Wv, Wk4, Wv4);
  k_addbias<<<(HWN * HH + 255) / 256, 256, 0, stream>>>(pos_emb, bimg, tmpPB);
  // posk/posv = (pos_emb+bimg) @ Wk/Wv + bk/bv   (M=3072, N=512, K=512) 16x64 tiles
  {
    const int tiles = (HWN / 16) * (HH / 64);  // 1536
    k_gemm4<<<(tiles + 7) / 8, 256, 0, stream>>>(tmpPB, HH, Wk, bk, posk, HWN, HH, 0);
    k_gemm4<<<(tiles + 7) / 8, 256, 0, stream>>>(tmpPB, HH, Wv, bv, posv, HWN, HH, 0);
  }
  // P0 = E @ Wproj[0:1536,:] + bproj   (M=128, N=2048, K=1536) 16x64 tiles
  {
    const int tiles = (TSTEPS * BB / 16) * (H4 / 64);  // 256
    k_gemm4<<<(tiles + 7) / 8, 256, 0, stream>>>(E, 1536, Wproj, bproj, P0,
                                                 TSTEPS * BB, H4, 0);
  }
  k_init<<<(BB * HH + 255) / 256, 256, 0, stream>>>(unk_h_lo, unk_c_lo, unk_h_up,
                                                    unk_c_up, fb_unk,
                                                    h_lo, c_lo, h_up, c_up, fb);

  const float* WprojFB = Wproj + (size_t)1536 * H4;  // rows 1536..2047

  for (int t = 0; t < TSTEPS; ++t) {
    const float* img_t = imgf + (size_t)t * BB * 4 * HWN;
    k_mask<<<(BB * HH + 255) / 256, 256, 0, stream>>>(padi + t * BB,
        pad_h_lo, pad_c_lo, pad_h_up, pad_c_up, fb_pad,
        h_lo, c_lo, h_up, c_up, fb);
    // proj = relu(P0[t] + fb @ Wproj[1536:,:])     (M=8, N=2048, K=512)
    k_gemm<<<16, 256, 0, stream>>>(fb, HH, WprojFB, nullptr, 0, nullptr,
                                   nullptr, P0 + (size_t)t * BB * H4,
                                   proj, BB, H4, 1);
    // g = proj @ Wih_lo + h_lo @ Whh_lo + b_lo     (M=8, N=2048, K=2048+512)
    k_gemm<<<16, 256, 0, stream>>>(proj, H4, Wih_lo, h_lo, HH, Whh_lo,
                                   b_lo, nullptr, g, BB, H4, 0);
    k_lstm<<<16, 256, 0, stream>>>(g, c_lo, h_lo, nullptr);
    // q = h_lo @ Wq + bq                           (M=8, N=512, K=512)
    k_gemm<<<4, 256, 0, stream>>>(h_lo, HH, Wq, nullptr, 0, nullptr,
                                  bq, nullptr, qb, BB, HH, 0);
    k_scores<<<BB * NHEAD, 256, 0, stream>>>(qb, img_t, posk, Wk4, sc);
    k_softmax<<<BB * NHEAD, 256, 0, stream>>>(sc);
    k_ctx<<<BB * NHEAD, 256, 0, stream>>>(sc, img_t, posv, Wv4, ctx);
    // ctxo = ctx @ Wo + bo + h_lo (residual)       (M=8, N=512, K=512)
    k_gemm<<<4, 256, 0, stream>>>(ctx, HH, Wo, nullptr, 0, nullptr,
                                  bo, h_lo, ctxo, BB, HH, 0);
    // g = ctxo @ Wih_up + h_up @ Whh_up + b_up     (M=8, N=2048, K=512+512)
    k_gemm<<<16, 256, 0, stream>>>(ctxo, HH, Wih_up, h_up, HH, Whh_up,
                                   b_up, nullptr, g, BB, H4, 0);
    k_lstm<<<16, 256, 0, stream>>>(g, c_up, h_up, fb);  // fb = h_up
  }

  // hid = relu(h_up @ Wout1 + bout1)               (M=8, N=2048, K=512)
  k_gemm<<<16, 256, 0, stream>>>(h_up, HH, Wout1, nullptr, 0, nullptr,
                                 bout1, nullptr, hid, BB, H4, 1);
  // out = hid @ Wout2 + bout2                      (M=8, N=49152, K=2048) 16x64 tiles
  {
    const int tiles = OUTN / 64;  // 768
    k_gemm4<<<(tiles + 7) / 8, 256, 0, stream>>>(hid, H4, Wout2, bout2,
                                                 (float*)d_out, BB, OUTN, 0);
  }
}